// InstanceSegmentationLoss_21285857919581
// MI455X (gfx1250) — compile-verified
//
#include <hip/hip_runtime.h>

// ---------------------------------------------------------------------------
// InstanceSegmentationLoss on MI455X (gfx1250)
//
// Strategy: the whole loss is a function of the joint 33x33 id histogram.
//   Phase 1: bandwidth-bound streaming histogram (async-LDS staged loads,
//            LDS-privatized ds_add_u32 bins, global-atomic flush into d_ws).
//            32 MiB @ 23.3 TB/s => ~1.4 us floor; ~3 VALU + 1 DS op/pixel.
//   Phase 2: single-wave finalize; marginal row/col sums of H via chained
//            v_wmma_f32_16x16x4_f32 (exact in f32), then per-lane IoU and
//            cross-lane max/sum reductions.
// ---------------------------------------------------------------------------

#define HW      (2048u * 2048u)
#define NINST   33u                 // ids 0..32 (0 = background)
#define NBINS   (NINST * NINST)     // 1089
#define BLOCK   256u
#define GRID    1024u
#define VEC4    (HW / 4u)           // 1,048,576 float4 per input
#define STRIDE  (BLOCK * GRID)      // 262,144
#define TILES   (VEC4 / STRIDE)     // 4 (exact)

#ifndef __has_builtin
#define __has_builtin(x) 0
#endif

// ---- CDNA5 async global->LDS staging (ASYNCcnt-tracked) -------------------
typedef int v4i __attribute__((ext_vector_type(4)));
typedef __attribute__((address_space(1))) v4i gv4i;  // global
typedef __attribute__((address_space(3))) v4i lv4i;  // LDS

#if __has_builtin(__builtin_amdgcn_global_load_async_to_lds_b128)
#define HAVE_ASYNC_LDS 1
#else
#define HAVE_ASYNC_LDS 0
#endif

#if __has_builtin(__builtin_amdgcn_s_wait_asynccnt)
#define WAIT_ASYNC(n) __builtin_amdgcn_s_wait_asynccnt(n)
#else
#define WAIT_ASYNC(n) asm volatile("s_wait_asynccnt %0" ::"n"(n) : "memory")
#endif

__device__ __forceinline__ void stage16(const float4* g, float4* l) {
#if HAVE_ASYNC_LDS
  // Signature (from clang diagnostic): (v4i AS1* src, v4i AS3* dst, Ii, Ii).
  // Integer round-trips: generic global ptr value == AS1 value; generic LDS
  // ptr low 32 bits == AS3 offset (ISA 10.2 aperture rules).
  gv4i* gs = (gv4i*)(unsigned long long)g;
  lv4i* ls = (lv4i*)(unsigned)(unsigned long long)l;
  __builtin_amdgcn_global_load_async_to_lds_b128(gs, ls, 0, 0);
#else
  *l = *g;  // fallback: synchronous copy through VGPRs
#endif
}

// ---------------------------------------------------------------------------
__global__ void zero_hist_kernel(unsigned* __restrict__ ghist) {
  unsigned i = blockIdx.x * blockDim.x + threadIdx.x;
  if (i < NBINS) ghist[i] = 0u;
}

// ---------------------------------------------------------------------------
__global__ void __launch_bounds__(BLOCK)
hist_kernel(const float* __restrict__ pred, const float* __restrict__ truem,
            unsigned* __restrict__ ghist) {
  __shared__ unsigned hist[NBINS];
  __shared__ float4 pbuf[2][BLOCK];
  __shared__ float4 tbuf[2][BLOCK];

  const unsigned tid = threadIdx.x;
  for (unsigned i = tid; i < NBINS; i += BLOCK) hist[i] = 0u;
  __syncthreads();

  const float4* p4 = (const float4*)pred;
  const float4* t4 = (const float4*)truem;
  const unsigned idx = blockIdx.x * BLOCK + tid;

  // prologue: stage tile 0
  stage16(p4 + idx, &pbuf[0][tid]);
  stage16(t4 + idx, &tbuf[0][tid]);

  for (unsigned t = 0; t < TILES; ++t) {
    if (t + 1u < TILES) {  // stage next tile, then wait for current (<=2 left)
      const unsigned nidx = idx + (t + 1u) * STRIDE;
      stage16(p4 + nidx, &pbuf[(t + 1u) & 1u][tid]);
      stage16(t4 + nidx, &tbuf[(t + 1u) & 1u][tid]);
      WAIT_ASYNC(2);
    } else {
      WAIT_ASYNC(0);
    }
    asm volatile("" ::: "memory");  // compiler barrier: reads below stay below

    const float4 p = pbuf[t & 1u][tid];
    const float4 q = tbuf[t & 1u][tid];
    atomicAdd(&hist[(unsigned)p.x * NINST + (unsigned)q.x], 1u);
    atomicAdd(&hist[(unsigned)p.y * NINST + (unsigned)q.y], 1u);
    atomicAdd(&hist[(unsigned)p.z * NINST + (unsigned)q.z], 1u);
    atomicAdd(&hist[(unsigned)p.w * NINST + (unsigned)q.w], 1u);
  }

  __syncthreads();
  for (unsigned i = tid; i < NBINS; i += BLOCK) {
    const unsigned c = hist[i];
    if (c) atomicAdd(&ghist[i], c);
  }
}

// ---------------------------------------------------------------------------
// Finalize: one wave (32 lanes).
typedef float v8f __attribute__((ext_vector_type(8)));
typedef float v2f __attribute__((ext_vector_type(2)));

#define HPAD (36u * 33u)  // p padded with zero rows to 36; cols stay 33 (row stride 33)

__global__ void __launch_bounds__(32)
finalize_kernel(const unsigned* __restrict__ ghist, float* __restrict__ out) {
  __shared__ float H[HPAD];   // H[p*33 + t], zero-padded to 36 in p
  __shared__ float Rsum[34];  // row sums   (psum over true ids 0..32)
  __shared__ float Csum[34];  // col sums   (tsum over pred ids 0..32)

  const int lane = threadIdx.x;  // 0..31, single wave
  for (unsigned i = lane; i < HPAD; i += 32u)
    H[i] = (i < NBINS) ? (float)ghist[i] : 0.0f;
  __syncthreads();

#if __has_builtin(__builtin_amdgcn_wmma_f32_16x16x4_f32)
  // Marginals on the matrix pipe: D = A(16x4) x B(4x16) + C, K chained 0..35.
  // A layout (ISA 7.12.2): lane m=l&15 -> M=m; VGPR v holds K = v + 2*(l>=16).
  // C/D layout: VGPR v, lane l -> M = v + 8*(l>=16), N = l&15.
  const int hi = lane >> 4;
  const int m  = lane & 15;

  // Row sums of rows 1..32: A = H-tile, B = ones. D[i][j] = rowsum(r0+i).
  for (int rt = 0; rt < 2; ++rt) {
    const int r0 = 1 + rt * 16;
    v8f acc = {};
    const v2f ones = {1.0f, 1.0f};
    for (int kk = 0; kk < 36; kk += 4) {
      v2f a;
      a.x = H[(r0 + m) * 33 + kk + 0 + 2 * hi];
      a.y = H[(r0 + m) * 33 + kk + 1 + 2 * hi];
      acc = __builtin_amdgcn_wmma_f32_16x16x4_f32(false, a, false, ones,
                                                  (short)0, acc, false, false);
    }
#pragma unroll
    for (int v = 0; v < 8; ++v) Rsum[r0 + v + 8 * hi] = acc[v];
  }

  // Col sums of cols 1..32: A = ones, B = H-tile. D[i][j] = colsum(c0+j).
  for (int ct = 0; ct < 2; ++ct) {
    const int c0 = 1 + ct * 16;
    v8f acc = {};
    const v2f ones = {1.0f, 1.0f};
    for (int kk = 0; kk < 36; kk += 4) {
      v2f b;
      b.x = H[(kk + 0 + 2 * hi) * 33 + c0 + m];
      b.y = H[(kk + 1 + 2 * hi) * 33 + c0 + m];
      acc = __builtin_amdgcn_wmma_f32_16x16x4_f32(false, ones, false, b,
                                                  (short)0, acc, false, false);
    }
    Csum[c0 + m] = acc[0];
  }
  if (lane == 0) { Rsum[0] = 0.f; Csum[0] = 0.f; }  // unused (weight 0)
#else
  for (int r = lane; r < 33; r += 32) {
    float s = 0.f;
    for (int k = 0; k < 33; ++k) s += H[r * 33 + k];
    Rsum[r] = s;
  }
  for (int c = lane; c < 33; c += 32) {
    float s = 0.f;
    for (int k = 0; k < 33; ++k) s += H[k * 33 + c];
    Csum[c] = s;
  }
#endif
  __syncthreads();

  // lane <-> true id (lane+1); loop n over pred ids.
  const float tsum = Csum[lane + 1];
  double loss = 0.0;
  for (int n = 0; n < 32; ++n) {
    const float pn    = Rsum[n + 1];
    const float inter = H[(n + 1) * 33 + (lane + 1)];
    const float uni   = pn + tsum - inter;
    float iou = (uni > 0.f) ? (inter / uni) : 0.f;
    for (int off = 16; off > 0; off >>= 1)
      iou = fmaxf(iou, __shfl_xor(iou, off, 32));
    loss += (double)(1.0f - fmaxf(iou, 0.f));
  }

  // dummy = (sum(pred) + sum(true)) / 1e12, from id-weighted marginals.
  const float psum = Rsum[lane + 1];
  double contrib = (double)(lane + 1) * ((double)psum + (double)tsum);
  for (int off = 16; off > 0; off >>= 1)
    contrib += __shfl_xor(contrib, off, 32);

  if (lane == 0) out[0] = (float)(loss + contrib * 1e-12);
}

// ---------------------------------------------------------------------------
extern "C" void kernel_launch(void* const* d_in, const int* in_sizes, int n_in,
                              void* d_out, int out_size, void* d_ws,
                              size_t ws_size, hipStream_t stream) {
  (void)in_sizes; (void)n_in; (void)out_size; (void)ws_size;
  const float* pred = (const float*)d_in[0];
  const float* tru  = (const float*)d_in[1];
  unsigned* ghist   = (unsigned*)d_ws;  // needs 1089 * 4 = 4356 bytes
  float* out        = (float*)d_out;

  zero_hist_kernel<<<dim3((NBINS + 255u) / 256u), dim3(256), 0, stream>>>(ghist);
  hist_kernel<<<dim3(GRID), dim3(BLOCK), 0, stream>>>(pred, tru, ghist);
  finalize_kernel<<<dim3(1), dim3(32), 0, stream>>>(ghist, out);
}